// EGNNLayer_32753420599912
// MI455X (gfx1250) — compile-verified
//
#include <hip/hip_runtime.h>
#include <hip/hip_bf16.h>
#include <math.h>

// Problem constants from the reference
#define BB 4
#define NN 512
#define DD 64
#define MM (BB * NN)   // 2048 flattened rows

typedef float v2f __attribute__((ext_vector_type(2)));
typedef float v8f __attribute__((ext_vector_type(8)));

__device__ __forceinline__ float silu_f(float x) {
    return x / (1.0f + expf(-x));
}
__device__ __forceinline__ float sigmoid_f(float x) {
    return 1.0f / (1.0f + expf(-x));
}

// ---------------------------------------------------------------------------
// Kernel 1: per-node dot products   a_i = h_i . eW1[0:64]   b_i = h_i . eW1[64:128]
// ---------------------------------------------------------------------------
__global__ void egnn_node_dots(const float* __restrict__ h,
                               const float* __restrict__ eW1,
                               float* __restrict__ Aout,
                               float* __restrict__ Bout) {
    int row = blockIdx.x * blockDim.x + threadIdx.x;
    if (row >= MM) return;
    const float* hr = h + (size_t)row * DD;
    float a = 0.0f, b = 0.0f;
    #pragma unroll 8
    for (int k = 0; k < DD; ++k) {
        float hv = hr[k];
        a = fmaf(hv, eW1[k], a);
        b = fmaf(hv, eW1[DD + k], b);
    }
    Aout[row] = a;
    Bout[row] = b;
}

// ---------------------------------------------------------------------------
// Kernel 2: pair sweep. One block per row (b,i). Streams rij/dij (memory bound),
// computes scalar edge chain, reduces x-update (3 floats) and S_i = sum_j t_ij.
// Writes x_new directly into d_out, S into workspace.
// ---------------------------------------------------------------------------
__global__ void egnn_pair_sweep(const float* __restrict__ Aw,
                                const float* __restrict__ Bw,
                                const float* __restrict__ dij,
                                const float* __restrict__ rij,
                                const float* __restrict__ mask,
                                const float* __restrict__ x,
                                const float* __restrict__ eW1,   // w_d = eW1[128]
                                const float* __restrict__ eb1p,
                                const float* __restrict__ eW2p,
                                const float* __restrict__ eb2p,
                                const float* __restrict__ eW3p,
                                const float* __restrict__ eb3p,
                                const float* __restrict__ pW1p,
                                const float* __restrict__ pb1p,
                                const float* __restrict__ pW2p,
                                const float* __restrict__ pb2p,
                                const float* __restrict__ nW1p,
                                const float* __restrict__ nb1p,
                                float* __restrict__ x_out,
                                float* __restrict__ S_out) {
    const int row  = blockIdx.x;          // b*N + i
    const int b    = row / NN;
    const int tid  = threadIdx.x;
    const int base = b * NN;

    const float ai    = Aw[row];
    const float maskI = mask[row];
    const float w_d = eW1[2 * DD];
    const float eb1 = eb1p[0];
    const float c2  = eW2p[0], d2 = eb2p[0];
    const float c3  = eW3p[0], d3 = eb3p[0];
    const float pw1 = pW1p[0], pb1 = pb1p[0];
    const float pw2 = pW2p[0], pb2 = pb2p[0];
    const float nw1 = nW1p[0], nb1 = nb1p[0];

    const float* __restrict__ drow = dij + (size_t)row * NN;
    const float* __restrict__ rrow = rij + (size_t)row * NN * 3;

    float xx = 0.0f, xy = 0.0f, xz = 0.0f, sS = 0.0f;

    for (int j = tid; j < NN; j += 256) {
        const float* rp = rrow + (size_t)j * 3;
        __builtin_prefetch(rp + 256 * 3, 0, 0);  // next stripe of the rij stream
        float bj = Bw[base + j];
        float d  = drow[j];
        float s  = ai + bj + fmaf(d, w_d, eb1);
        float m1 = silu_f(s);
        float m2 = silu_f(fmaf(m1, c2, d2));
        float mm = fmaf(m2, c3, d3) * maskI * mask[base + j];
        // phi_x scalar chain
        float ph = fmaf(silu_f(fmaf(mm, pw1, pb1)), pw2, pb2);
        xx = fmaf(rp[0], ph, xx);
        xy = fmaf(rp[1], ph, xy);
        xz = fmaf(rp[2], ph, xz);
        // msg_to_node first stage (scalar); second stage factored out
        sS += silu_f(fmaf(mm, nw1, nb1));
    }

    __shared__ float sX[256], sY[256], sZ[256], sSm[256];
    sX[tid] = xx; sY[tid] = xy; sZ[tid] = xz; sSm[tid] = sS;
    __syncthreads();
    for (int st = 128; st > 0; st >>= 1) {
        if (tid < st) {
            sX[tid] += sX[tid + st];
            sY[tid] += sY[tid + st];
            sZ[tid] += sZ[tid + st];
            sSm[tid] += sSm[tid + st];
        }
        __syncthreads();
    }
    if (tid == 0) {
        x_out[(size_t)row * 3 + 0] = x[(size_t)row * 3 + 0] + sX[0];
        x_out[(size_t)row * 3 + 1] = x[(size_t)row * 3 + 1] + sY[0];
        x_out[(size_t)row * 3 + 2] = x[(size_t)row * 3 + 2] + sZ[0];
        S_out[row] = sSm[0];
    }
}

// ---------------------------------------------------------------------------
// Kernel 3: fused GRU via V_WMMA_F32_16X16X4_F32.
//   m_node[m][k] = S[m]*nW2[k] + N*nb2[k]   (synthesized as A fragments)
//   gi = m_node @ Wih^T ; gh = h @ Whh^T    (six 16x16 accumulators per wave)
//   gates + mask + store h_new, all in C/D register layout.
// One wave per (16-row, 16-col) tile: 128 row-tiles x 4 col-tiles = 512 waves.
// ---------------------------------------------------------------------------
__global__ void egnn_gru_wmma(const float* __restrict__ h,
                              const float* __restrict__ Sws,
                              const float* __restrict__ nW2,
                              const float* __restrict__ nb2,
                              const float* __restrict__ Wih,
                              const float* __restrict__ bih,
                              const float* __restrict__ Whh,
                              const float* __restrict__ bhh,
                              const float* __restrict__ mask,
                              float* __restrict__ h_out) {
    const int wave = (blockIdx.x * blockDim.x + threadIdx.x) >> 5;  // wave32
    const int lane = threadIdx.x & 31;
    const int rt = wave >> 2;        // 0..127 row tile
    const int ct = wave & 3;         // 0..3   col tile (of D=64)
    const int m0 = rt << 4;
    const int d0 = ct << 4;
    const bool hi = lane >= 16;
    const int l15 = lane & 15;
    const int mrow = m0 + l15;       // A-fragment row owned by this lane
    const int ncol = d0 + l15;       // B/C/D column owned by this lane

    const float Srow = Sws[mrow];
    const float* __restrict__ hrow = h + (size_t)mrow * DD;

    v8f air = {}, aiz = {}, ain = {};
    v8f ahr = {}, ahz = {}, ahn = {};

    const int nr = ncol;             // Wih/Whh row indices for r/z/n gate blocks
    const int nz = DD + ncol;
    const int nn2 = 2 * DD + ncol;
    const float* __restrict__ Wir = Wih + (size_t)nr * DD;
    const float* __restrict__ Wiz = Wih + (size_t)nz * DD;
    const float* __restrict__ Win = Wih + (size_t)nn2 * DD;
    const float* __restrict__ Whr = Whh + (size_t)nr * DD;
    const float* __restrict__ Whz = Whh + (size_t)nz * DD;
    const float* __restrict__ Whn = Whh + (size_t)nn2 * DD;

    #pragma unroll
    for (int k0 = 0; k0 < DD; k0 += 4) {
        const int kk = k0 + (hi ? 2 : 0);
        // A fragment: m_node row (synthesized) and h row
        v2f am, ah;
        am.x = fmaf(Srow, nW2[kk],     (float)NN * nb2[kk]);
        am.y = fmaf(Srow, nW2[kk + 1], (float)NN * nb2[kk + 1]);
        ah.x = hrow[kk];
        ah.y = hrow[kk + 1];
        // B fragments: columns of Wih^T / Whh^T == rows of Wih / Whh
        v2f bir, biz, bin, bhr, bhz, bhn;
        bir.x = Wir[kk]; bir.y = Wir[kk + 1];
        biz.x = Wiz[kk]; biz.y = Wiz[kk + 1];
        bin.x = Win[kk]; bin.y = Win[kk + 1];
        bhr.x = Whr[kk]; bhr.y = Whr[kk + 1];
        bhz.x = Whz[kk]; bhz.y = Whz[kk + 1];
        bhn.x = Whn[kk]; bhn.y = Whn[kk + 1];

        air = __builtin_amdgcn_wmma_f32_16x16x4_f32(false, am, false, bir, (short)0, air, false, false);
        aiz = __builtin_amdgcn_wmma_f32_16x16x4_f32(false, am, false, biz, (short)0, aiz, false, false);
        ain = __builtin_amdgcn_wmma_f32_16x16x4_f32(false, am, false, bin, (short)0, ain, false, false);
        ahr = __builtin_amdgcn_wmma_f32_16x16x4_f32(false, ah, false, bhr, (short)0, ahr, false, false);
        ahz = __builtin_amdgcn_wmma_f32_16x16x4_f32(false, ah, false, bhz, (short)0, ahz, false, false);
        ahn = __builtin_amdgcn_wmma_f32_16x16x4_f32(false, ah, false, bhn, (short)0, ahn, false, false);
    }

    // Per-lane biases (depend only on the column this lane owns)
    const float bir_b = bih[nr], biz_b = bih[nz], bin_b = bih[nn2];
    const float bhr_b = bhh[nr], bhz_b = bhh[nz], bhn_b = bhh[nn2];

    #pragma unroll
    for (int r = 0; r < 8; ++r) {
        const int row = m0 + r + (hi ? 8 : 0);   // C/D layout: M = r + 8*half
        float rg = sigmoid_f(air[r] + bir_b + ahr[r] + bhr_b);
        float zg = sigmoid_f(aiz[r] + biz_b + ahz[r] + bhz_b);
        float ng = tanhf(ain[r] + bin_b + rg * (ahn[r] + bhn_b));
        float hold = h[(size_t)row * DD + ncol];
        float hv = (1.0f - zg) * ng + zg * hold;
        hv *= mask[row];
        h_out[(size_t)row * DD + ncol] = hv;
    }
}

// ---------------------------------------------------------------------------
extern "C" void kernel_launch(void* const* d_in, const int* in_sizes, int n_in,
                              void* d_out, int out_size, void* d_ws, size_t ws_size,
                              hipStream_t stream) {
    const float* h    = (const float*)d_in[0];
    const float* x    = (const float*)d_in[1];
    const float* rij  = (const float*)d_in[2];
    const float* dij  = (const float*)d_in[3];
    const float* mask = (const float*)d_in[4];
    const float* eW1  = (const float*)d_in[5];
    const float* eb1  = (const float*)d_in[6];
    const float* eW2  = (const float*)d_in[7];
    const float* eb2  = (const float*)d_in[8];
    const float* eW3  = (const float*)d_in[9];
    const float* eb3  = (const float*)d_in[10];
    const float* pW1  = (const float*)d_in[11];
    const float* pb1  = (const float*)d_in[12];
    const float* pW2  = (const float*)d_in[13];
    const float* pb2  = (const float*)d_in[14];
    const float* nW1  = (const float*)d_in[15];
    const float* nb1  = (const float*)d_in[16];
    const float* nW2  = (const float*)d_in[17];
    const float* nb2  = (const float*)d_in[18];
    const float* Wih  = (const float*)d_in[19];
    const float* bih  = (const float*)d_in[20];
    const float* Whh  = (const float*)d_in[21];
    const float* bhh  = (const float*)d_in[22];

    float* h_out = (float*)d_out;                 // (B,N,D) flat = 131072
    float* x_out = h_out + (size_t)MM * DD;       // (B,N,3)  flat = 6144

    float* ws  = (float*)d_ws;
    float* Aw  = ws;            // 2048
    float* Bw  = ws + MM;       // 2048
    float* Sws = ws + 2 * MM;   // 2048

    // K1: per-node dots
    egnn_node_dots<<<MM / 256, 256, 0, stream>>>(h, eW1, Aw, Bw);

    // K2: pair sweep (memory-bound over rij/dij), writes x_new + S
    egnn_pair_sweep<<<MM, 256, 0, stream>>>(Aw, Bw, dij, rij, mask, x,
                                            eW1, eb1, eW2, eb2, eW3, eb3,
                                            pW1, pb1, pW2, pb2, nW1, nb1,
                                            x_out, Sws);

    // K3: fused GRU via WMMA. 512 tiles, 8 waves/block of 256 threads -> 64 blocks
    egnn_gru_wmma<<<64, 256, 0, stream>>>(h, Sws, nW2, nb2, Wih, bih, Whh, bhh,
                                          mask, h_out);
}